// GATv2_68006512165292
// MI455X (gfx1250) — compile-verified
//
#include <hip/hip_runtime.h>
#include <math.h>

typedef float v2f __attribute__((ext_vector_type(2)));
typedef float v8f __attribute__((ext_vector_type(8)));

#define K_NEG_SLOPE 0.2f

// ---------- monotone float<->uint encoding for atomic segment-max ----------
__device__ __forceinline__ unsigned int f32_ord(float f) {
  unsigned int u = __float_as_uint(f);
  return (u & 0x80000000u) ? ~u : (u | 0x80000000u);
}
__device__ __forceinline__ float ord_f32(unsigned int u) {
  return (u & 0x80000000u) ? __uint_as_float(u ^ 0x80000000u)
                           : __uint_as_float(~u);
}
// encoding of -inf: ~0xFF800000 = 0x007FFFFF
#define ORD_NEG_INF 0x007FFFFFu

// ------------------------------- fills ------------------------------------
__global__ void fill_f32_k(float* __restrict__ p, float v, long long n) {
  long long i = (long long)blockIdx.x * blockDim.x + threadIdx.x;
  long long st = (long long)gridDim.x * blockDim.x;
  for (; i < n; i += st) p[i] = v;
}
__global__ void fill_u32_k(unsigned int* __restrict__ p, unsigned int v, long long n) {
  long long i = (long long)blockIdx.x * blockDim.x + threadIdx.x;
  long long st = (long long)gridDim.x * blockDim.x;
  for (; i < n; i += st) p[i] = v;
}

// --------------------- GEMM: Y[rows x 128] = X @ W + b ---------------------
// One block = 64 output rows. 8 waves, wave w owns cols [16w,16w+16) and all
// four 16-row M-tiles: each B fragment is loaded once per k-step and reused
// across 4x V_WMMA_F32_16X16X4_F32 (fp32-exact MMA).
__global__ __launch_bounds__(256) void gemm_nk128_bias(
    const float* __restrict__ X, const float* __restrict__ W,
    const float* __restrict__ bias, float* __restrict__ Y, int rows) {
  __shared__ float Xs[64][132];  // +4 pad: 132 % 64 = 4 -> conflict-free column reads
  const int row0 = blockIdx.x * 64;
  const int tid = threadIdx.x;

  // cooperative load of the 64x128 A-strip (2048 float4, 8 per thread)
  for (int i = tid; i < 2048; i += 256) {
    int r = i >> 5;           // 32 float4 per row
    int c4 = (i & 31) * 4;
    float4 v = make_float4(0.f, 0.f, 0.f, 0.f);
    if (row0 + r < rows)
      v = *(const float4*)(X + (long long)(row0 + r) * 128 + c4);
    Xs[r][c4 + 0] = v.x; Xs[r][c4 + 1] = v.y;
    Xs[r][c4 + 2] = v.z; Xs[r][c4 + 3] = v.w;
  }
  __syncthreads();

  const int lane = tid & 31;
  const int wave = tid >> 5;
  const int n0   = wave * 16;
  const int mrow = lane & 15;     // A row within M-tile this lane holds
  const int half = lane >> 4;     // 0: K0/K1, 1: K2/K3 (ISA 16x4 f32 layout)
  const int ncol = n0 + (lane & 15);

  v8f acc0 = {}, acc1 = {}, acc2 = {}, acc3 = {};
#pragma unroll
  for (int k = 0; k < 128; k += 4) {
    v2f b;
    b.x = W[(long long)(k + half * 2 + 0) * 128 + ncol];
    b.y = W[(long long)(k + half * 2 + 1) * 128 + ncol];
    v2f a0, a1, a2, a3;
    a0.x = Xs[mrow +  0][k + half * 2 + 0]; a0.y = Xs[mrow +  0][k + half * 2 + 1];
    a1.x = Xs[mrow + 16][k + half * 2 + 0]; a1.y = Xs[mrow + 16][k + half * 2 + 1];
    a2.x = Xs[mrow + 32][k + half * 2 + 0]; a2.y = Xs[mrow + 32][k + half * 2 + 1];
    a3.x = Xs[mrow + 48][k + half * 2 + 0]; a3.y = Xs[mrow + 48][k + half * 2 + 1];
    acc0 = __builtin_amdgcn_wmma_f32_16x16x4_f32(false, a0, false, b, (short)0, acc0, false, false);
    acc1 = __builtin_amdgcn_wmma_f32_16x16x4_f32(false, a1, false, b, (short)0, acc1, false, false);
    acc2 = __builtin_amdgcn_wmma_f32_16x16x4_f32(false, a2, false, b, (short)0, acc2, false, false);
    acc3 = __builtin_amdgcn_wmma_f32_16x16x4_f32(false, a3, false, b, (short)0, acc3, false, false);
  }

  const float bv = bias[ncol];
  // C/D layout: VGPR j -> M = j + 8*half within each 16-row tile
#pragma unroll
  for (int j = 0; j < 8; ++j) {
    int m = row0 + j + half * 8;
    if (m      < rows) Y[(long long)(m     ) * 128 + ncol] = acc0[j] + bv;
    if (m + 16 < rows) Y[(long long)(m + 16) * 128 + ncol] = acc1[j] + bv;
    if (m + 32 < rows) Y[(long long)(m + 32) * 128 + ncol] = acc2[j] + bv;
    if (m + 48 < rows) Y[(long long)(m + 48) * 128 + ncol] = acc3[j] + bv;
  }
}

// ------------- edge scores + segment-max (one wave per edge) ---------------
// lane l covers channels [4l, 4l+4); for H heads, 32/H lanes form one head.
template <int H>
__global__ __launch_bounds__(256) void edge_scores_k(
    const float* __restrict__ xl, const float* __restrict__ xr,
    const int* __restrict__ src, const int* __restrict__ dst,
    const float* __restrict__ att, float* __restrict__ score,
    unsigned int* __restrict__ mord, int Eorig, int Etot) {
  int wid = blockIdx.x * (blockDim.x >> 5) + (threadIdx.x >> 5);
  if (wid >= Etot) return;
  const int lane = threadIdx.x & 31;
  int s, d;
  if (wid < Eorig) { s = src[wid]; d = dst[wid]; }
  else             { s = d = wid - Eorig; }

  float4 va = *(const float4*)(xl + (long long)s * 128 + lane * 4);
  float4 vb = *(const float4*)(xr + (long long)d * 128 + lane * 4);
  float4 at = *(const float4*)(att + lane * 4);

  float e0 = va.x + vb.x; e0 = e0 > 0.f ? e0 : K_NEG_SLOPE * e0;
  float e1 = va.y + vb.y; e1 = e1 > 0.f ? e1 : K_NEG_SLOPE * e1;
  float e2 = va.z + vb.z; e2 = e2 > 0.f ? e2 : K_NEG_SLOPE * e2;
  float e3 = va.w + vb.w; e3 = e3 > 0.f ? e3 : K_NEG_SLOPE * e3;
  float p = e0 * at.x + e1 * at.y + e2 * at.z + e3 * at.w;

  const int span = 32 / H;
#pragma unroll
  for (int off = 1; off < span; off <<= 1) p += __shfl_xor(p, off);

  if ((lane & (span - 1)) == 0) {
    int h = lane / span;
    score[(long long)wid * H + h] = p;
    atomicMax(&mord[(long long)d * H + h], f32_ord(p));
  }
}

// --------------- p = exp(score - m[dst]); denom += p -----------------------
template <int H>
__global__ void softmax_denom_k(
    float* __restrict__ score, const unsigned int* __restrict__ mord,
    const int* __restrict__ dst, float* __restrict__ denom,
    int Eorig, int Etot) {
  long long i = (long long)blockIdx.x * blockDim.x + threadIdx.x;
  if (i >= (long long)Etot * H) return;
  int e = (int)(i / H);
  int h = (int)(i - (long long)e * H);
  int d = (e < Eorig) ? dst[e] : (e - Eorig);
  float m = ord_f32(mord[(long long)d * H + h]);
  float p = expf(score[i] - m);
  score[i] = p;  // overwrite score with numerator
  atomicAdd(&denom[(long long)d * H + h], p);
}

// ------------ out[dst] += (p/denom[dst]) * xl[src] (wave/edge) -------------
template <int H>
__global__ __launch_bounds__(256) void edge_aggregate_k(
    const float* __restrict__ p, const float* __restrict__ denom,
    const float* __restrict__ xl,
    const int* __restrict__ src, const int* __restrict__ dst,
    float* __restrict__ acc, int Eorig, int Etot) {
  int wid = blockIdx.x * (blockDim.x >> 5) + (threadIdx.x >> 5);
  if (wid >= Etot) return;
  const int lane = threadIdx.x & 31;
  int s, d;
  if (wid < Eorig) { s = src[wid]; d = dst[wid]; }
  else             { s = d = wid - Eorig; }

  const int h = (lane * 4) / (128 / H);
  float alpha = p[(long long)wid * H + h] / denom[(long long)d * H + h];
  float4 v = *(const float4*)(xl + (long long)s * 128 + lane * 4);
  float* o = acc + (long long)d * 128 + lane * 4;
  atomicAdd(o + 0, alpha * v.x);
  atomicAdd(o + 1, alpha * v.y);
  atomicAdd(o + 2, alpha * v.z);
  atomicAdd(o + 3, alpha * v.w);
}

// ------------------------- out = elu(in + bias[c]) -------------------------
__global__ void bias_elu_k(const float* __restrict__ in,
                           const float* __restrict__ bias,
                           float* __restrict__ out, long long n) {
  long long i = (long long)blockIdx.x * blockDim.x + threadIdx.x;
  if (i >= n) return;
  float v = in[i] + bias[i & 127];
  out[i] = v > 0.f ? v : (expf(v) - 1.0f);
}

// ------------------------- global mean pool (atomics) ----------------------
__global__ void pool_sum_k(const float* __restrict__ h,
                           const int* __restrict__ batch,
                           float* __restrict__ sums, float* __restrict__ cnt,
                           int Nn) {
  long long i = (long long)blockIdx.x * blockDim.x + threadIdx.x;
  if (i >= (long long)Nn * 128) return;
  int n = (int)(i >> 7);
  int c = (int)(i & 127);
  int b = batch[n];
  atomicAdd(&sums[(long long)b * 128 + c], h[i]);
  if (c == 0) atomicAdd(&cnt[b], 1.0f);
}

// --------------------- pooled @ W3 + b3  (64 x 10) -------------------------
__global__ void final_linear_k(const float* __restrict__ sums,
                               const float* __restrict__ cnt,
                               const float* __restrict__ W3,
                               const float* __restrict__ b3,
                               float* __restrict__ out, int G) {
  int g = blockIdx.x;
  int j = threadIdx.x;
  if (g >= G || j >= 10) return;
  float c = cnt[g];
  c = c > 1.0f ? c : 1.0f;
  float s = b3[j];
  for (int k = 0; k < 128; ++k)
    s += (sums[(long long)g * 128 + k] / c) * W3[k * 10 + j];
  out[g * 10 + j] = s;
}

// ===========================================================================
extern "C" void kernel_launch(void* const* d_in, const int* in_sizes, int n_in,
                              void* d_out, int out_size, void* d_ws, size_t ws_size,
                              hipStream_t stream) {
  const float* x     = (const float*)d_in[0];
  const int*   ei    = (const int*)d_in[1];
  const int*   batch = (const int*)d_in[2];
  const float* W1_l  = (const float*)d_in[3];
  const float* b1_l  = (const float*)d_in[4];
  const float* W1_r  = (const float*)d_in[5];
  const float* b1_r  = (const float*)d_in[6];
  const float* att1  = (const float*)d_in[7];
  const float* bias1 = (const float*)d_in[8];
  const float* W2_l  = (const float*)d_in[9];
  const float* b2_l  = (const float*)d_in[10];
  const float* W2_r  = (const float*)d_in[11];
  const float* b2_r  = (const float*)d_in[12];
  const float* att2  = (const float*)d_in[13];
  const float* bias2 = (const float*)d_in[14];
  const float* W3    = (const float*)d_in[15];
  const float* b3    = (const float*)d_in[16];
  float* out = (float*)d_out;

  const int Nn   = in_sizes[0] / 128;
  const int E    = in_sizes[1] / 2;
  const int Etot = E + Nn;           // + self loops
  const int G    = out_size / 10;
  const int* src = ei;
  const int* dst = ei + E;

  // ---- workspace layout (floats) ----
  float* ws = (float*)d_ws;
  size_t o = 0;
  float* xl    = ws + o; o += (size_t)Nn * 128;
  float* xr    = ws + o; o += (size_t)Nn * 128;
  float* score = ws + o; o += (size_t)Etot * 8;
  unsigned int* mord = (unsigned int*)(ws + o); o += (size_t)Nn * 8;
  float* denom = ws + o; o += (size_t)Nn * 8;
  float* h1    = ws + o; o += (size_t)Nn * 128;
  float* acc   = ws + o; o += (size_t)Nn * 128;
  float* sums  = ws + o; o += (size_t)G * 128;
  float* cnt   = ws + o; o += (size_t)G;
  (void)ws_size; (void)n_in;

  const int gemmGrid = (Nn + 63) / 64;
  const int edgeGrid = (Etot + 7) / 8;            // 8 waves (edges) per block
  const long long NnD = (long long)Nn * 128;

  // ================= conv1 (H=8, C=16, concat) =================
  gemm_nk128_bias<<<gemmGrid, 256, 0, stream>>>(x, W1_l, b1_l, xl, Nn);
  gemm_nk128_bias<<<gemmGrid, 256, 0, stream>>>(x, W1_r, b1_r, xr, Nn);

  fill_u32_k<<<1024, 256, 0, stream>>>(mord, ORD_NEG_INF, (long long)Nn * 8);
  fill_f32_k<<<1024, 256, 0, stream>>>(denom, 0.f, (long long)Nn * 8);
  fill_f32_k<<<2048, 256, 0, stream>>>(acc, 0.f, NnD);

  edge_scores_k<8><<<edgeGrid, 256, 0, stream>>>(xl, xr, src, dst, att1,
                                                 score, mord, E, Etot);
  {
    long long n = (long long)Etot * 8;
    softmax_denom_k<8><<<(int)((n + 255) / 256), 256, 0, stream>>>(
        score, mord, dst, denom, E, Etot);
  }
  edge_aggregate_k<8><<<edgeGrid, 256, 0, stream>>>(score, denom, xl, src, dst,
                                                    acc, E, Etot);
  bias_elu_k<<<(int)((NnD + 255) / 256), 256, 0, stream>>>(acc, bias1, h1, NnD);

  // ================= conv2 (H=1, C=128, mean) ==================
  gemm_nk128_bias<<<gemmGrid, 256, 0, stream>>>(h1, W2_l, b2_l, xl, Nn);
  gemm_nk128_bias<<<gemmGrid, 256, 0, stream>>>(h1, W2_r, b2_r, xr, Nn);

  fill_u32_k<<<512, 256, 0, stream>>>(mord, ORD_NEG_INF, (long long)Nn);
  fill_f32_k<<<512, 256, 0, stream>>>(denom, 0.f, (long long)Nn);
  fill_f32_k<<<2048, 256, 0, stream>>>(acc, 0.f, NnD);

  edge_scores_k<1><<<edgeGrid, 256, 0, stream>>>(xl, xr, src, dst, att2,
                                                 score, mord, E, Etot);
  softmax_denom_k<1><<<(Etot + 255) / 256, 256, 0, stream>>>(
      score, mord, dst, denom, E, Etot);
  edge_aggregate_k<1><<<edgeGrid, 256, 0, stream>>>(score, denom, xl, src, dst,
                                                    acc, E, Etot);
  bias_elu_k<<<(int)((NnD + 255) / 256), 256, 0, stream>>>(acc, bias2, h1, NnD);

  // ================= pool + linear =============================
  fill_f32_k<<<64, 256, 0, stream>>>(sums, 0.f, (long long)G * 128 + G);
  pool_sum_k<<<(int)((NnD + 255) / 256), 256, 0, stream>>>(h1, batch, sums, cnt, Nn);
  final_linear_k<<<G, 32, 0, stream>>>(sums, cnt, W3, b3, out, G);
}